// GraphLoss_74371653697738
// MI455X (gfx1250) — compile-verified
//
#include <hip/hip_runtime.h>
#include <math.h>

// Problem constants (from reference)
#define BB  16
#define NN  512
#define MM  256
#define NCC 3

typedef float v2f __attribute__((ext_vector_type(2)));
typedef float v8f __attribute__((ext_vector_type(8)));

// ---------------------------------------------------------------------------
// init workspace: min-dist accumulators to +inf bits, msub contributions to 0
// ---------------------------------------------------------------------------
__global__ void init_ws_kernel(unsigned* __restrict__ minBits, float* __restrict__ msub) {
  int i = blockIdx.x * blockDim.x + threadIdx.x;
  if (i < BB * MM) { minBits[i] = 0x7F800000u; msub[i] = 0.0f; }
}

// ---------------------------------------------------------------------------
// Cost matrix via V_WMMA_F32_16X16X4_F32.
// dist^2 = |p|^2 - 2 p.g + |g|^2 folded into one K=4 WMMA:
//   A row n  = (x_n, y_n, |p_n|^2, 1)        (K=0,1 lanes 0-15; K=2,3 lanes 16-31)
//   B col m  = (-2 gx_m, -2 gy_m, 1, |g_m|^2)
// D[n,m] = squared normalized distance. Then cost_t[b,m,n] = 5*dist - sem[b,t_m,n].
// One wave (32 threads) = one 16(n) x 16(m) tile.  EXEC all-1s, no divergence
// around the WMMA as the ISA requires.
// ---------------------------------------------------------------------------
__global__ __launch_bounds__(32)
void cost_wmma_kernel(const float* __restrict__ positions,   // [B,N,2]
                      const float* __restrict__ semantics,   // [B,NC+1,N]
                      const float* __restrict__ pts_gt,      // [B,M,2]
                      const int*   __restrict__ pts_type,    // [B,M]
                      float* __restrict__ cost_t,            // [B,M,N]
                      unsigned* __restrict__ minBits)        // [B,M]
{
  const int lane = threadIdx.x;
  const int l    = lane & 15;
  const bool hi  = (lane & 16) != 0;
  const int b  = blockIdx.z;
  const int m0 = blockIdx.x * 16;
  const int n0 = blockIdx.y * 16;

  // normalized predicted vertex for this lane's A row
  const int n_row = n0 + l;
  const float px = positions[((size_t)b * NN + n_row) * 2 + 0] * (1.0f / 399.0f);
  const float py = positions[((size_t)b * NN + n_row) * 2 + 1] * (1.0f / 199.0f);
  v2f A;
  if (!hi) { A.x = px;              A.y = py;   }     // K=0,1
  else     { A.x = px * px + py * py; A.y = 1.0f; }   // K=2,3

  // normalized GT point for this lane's B column
  const int m_col = m0 + l;
  const float gx = (pts_gt[((size_t)b * MM + m_col) * 2 + 0] + 30.0f) * (1.0f / 60.0f);
  const float gy = (pts_gt[((size_t)b * MM + m_col) * 2 + 1] + 15.0f) * (1.0f / 30.0f);
  v2f Bv;
  if (!hi) { Bv.x = -2.0f * gx; Bv.y = -2.0f * gy; }  // K=0,1
  else     { Bv.x = 1.0f;       Bv.y = gx * gx + gy * gy; } // K=2,3

  v8f Cz = {};
  v8f D = __builtin_amdgcn_wmma_f32_16x16x4_f32(
      /*neg_a=*/false, A, /*neg_b=*/false, Bv,
      /*c_mod=*/(short)0, Cz, /*reuse_a=*/false, /*reuse_b=*/false);

  // D layout: VGPR r, lanes 0-15 -> row n0+r, col m0+lane;
  //           lanes 16-31 -> row n0+8+r, col m0+lane-16.
  const int m = m0 + l;                 // this lane's column (GT point)
  const int t = pts_type[b * MM + m];
  const int nbase = n0 + (hi ? 8 : 0);
  const float* __restrict__ semp = semantics + ((size_t)b * (NCC + 1) + t) * NN + nbase;
  float* __restrict__ crow = cost_t + ((size_t)b * MM + m) * NN + nbase;

  float dmin = 3.4e38f;
#pragma unroll
  for (int r = 0; r < 8; ++r) {
    float d2   = D[r];
    float dist = sqrtf(fmaxf(d2, 0.0f));
    dmin = fminf(dmin, dist);
    crow[r] = 5.0f * dist - semp[r];    // COST_DIST*dist + COST_CLASS*(-logp)
  }
  // non-negative float -> uint bit pattern is order preserving
  atomicMin(&minBits[b * MM + m], __float_as_uint(dmin));
}

// ---------------------------------------------------------------------------
// Jonker-Volgenant shortest augmenting path LAP, one workgroup per batch.
// cost: [B, R=MM rows (GT pts), C=NN cols (vertices)] in global (L2-resident).
// All solver state in LDS. Tie-break argmin by lowest index (matches np.argmin).
// ---------------------------------------------------------------------------
__global__ __launch_bounds__(256)
void hungarian_kernel(const float* __restrict__ cost, int* __restrict__ predOut) {
  const int R = MM, C = NN;
  const int tid = threadIdx.x;
  const int b = blockIdx.x;
  const float* __restrict__ cb = cost + (size_t)b * R * C;

  __shared__ float u[MM];
  __shared__ float v[NN];
  __shared__ float shortest[NN];
  __shared__ int   spred[NN];
  __shared__ int   row4col[NN];
  __shared__ int   col4row[MM];
  __shared__ unsigned char visited[NN];
  __shared__ float rval[256];
  __shared__ int   ridx[256];
  __shared__ float s_minval;
  __shared__ int   s_i, s_sink;

  for (int j = tid; j < C; j += 256) { v[j] = 0.0f; row4col[j] = -1; }
  for (int r = tid; r < R; r += 256) { u[r] = 0.0f; col4row[r] = -1; }
  __syncthreads();

  for (int cur = 0; cur < R; ++cur) {
    for (int j = tid; j < C; j += 256) { shortest[j] = 3.0e38f; spred[j] = -1; visited[j] = 0; }
    if (tid == 0) { s_i = cur; s_minval = 0.0f; s_sink = -1; }
    __syncthreads();

    while (true) {
      const int   i      = s_i;
      const float minval = s_minval;
      const float ui     = u[i];
      const float* __restrict__ crow = cb + (size_t)i * C;
      for (int j = tid; j < C; j += 256) {
        if (!visited[j]) {
          float d = minval + crow[j] - ui - v[j];
          if (d < shortest[j]) { shortest[j] = d; spred[j] = i; }
        }
      }
      __syncthreads();

      // block argmin over unvisited columns, prefer lowest index on ties
      float bv = 3.4e38f; int bj = C;
      for (int j = tid; j < C; j += 256) {
        float val = visited[j] ? 3.4e38f : shortest[j];
        if (val < bv || (val == bv && j < bj)) { bv = val; bj = j; }
      }
      rval[tid] = bv; ridx[tid] = bj;
      __syncthreads();
      for (int s = 128; s > 0; s >>= 1) {
        if (tid < s) {
          float ov = rval[tid + s]; int oj = ridx[tid + s];
          if (ov < rval[tid] || (ov == rval[tid] && oj < ridx[tid])) {
            rval[tid] = ov; ridx[tid] = oj;
          }
        }
        __syncthreads();
      }
      if (tid == 0) {
        int j = ridx[0];
        s_minval = rval[0];
        visited[j] = 1;
        if (row4col[j] < 0) s_sink = j; else s_i = row4col[j];
      }
      __syncthreads();
      if (s_sink >= 0) break;
    }

    // dual updates (row4col still pre-augmentation; distinct rows -> no races)
    const int   sink   = s_sink;
    const float minval = s_minval;
    for (int j = tid; j < C; j += 256) {
      if (visited[j]) {
        if (j != sink) u[row4col[j]] += minval - shortest[j];
        v[j] += shortest[j] - minval;
      }
    }
    if (tid == 0) u[cur] += minval;
    __syncthreads();

    // augment along predecessor chain (serial, thread 0)
    if (tid == 0) {
      int j = sink;
      while (j != -1) {
        int i2 = spred[j];
        row4col[j] = i2;
        int nx = col4row[i2];
        col4row[i2] = j;
        j = nx;
      }
    }
    __syncthreads();
  }

  for (int r = tid; r < R; r += 256) predOut[b * MM + r] = col4row[r];
}

// ---------------------------------------------------------------------------
// zero matches_gt region; default semantics_gt = NC (background) as float
// ---------------------------------------------------------------------------
__global__ void zero_out_kernel(float* __restrict__ out) {
  const size_t BNN   = (size_t)BB * NN * NN;
  const size_t total = BNN + (size_t)BB * NN;
  const size_t stride = (size_t)gridDim.x * blockDim.x;
  for (size_t i = (size_t)blockIdx.x * blockDim.x + threadIdx.x; i < total; i += stride)
    out[3 + i] = (i < BNN) ? 0.0f : (float)NCC;
}

// ---------------------------------------------------------------------------
// sum softplus(matches) -> one double partial per block (deterministic)
// ---------------------------------------------------------------------------
__global__ __launch_bounds__(256)
void softplus_kernel(const float* __restrict__ matches, double* __restrict__ partial) {
  __shared__ double sdata[256];
  const int n = BB * NN * NN;
  double s = 0.0;
  for (int i = blockIdx.x * blockDim.x + threadIdx.x; i < n; i += gridDim.x * blockDim.x) {
    float x = matches[i];
    s += (double)(fmaxf(x, 0.0f) + log1pf(expf(-fabsf(x))));  // stable softplus
  }
  sdata[threadIdx.x] = s;
  __syncthreads();
  for (int st = 128; st > 0; st >>= 1) {
    if (threadIdx.x < st) sdata[threadIdx.x] += sdata[threadIdx.x + st];
    __syncthreads();
  }
  if (threadIdx.x == 0) partial[blockIdx.x] = sdata[0];
}

// ---------------------------------------------------------------------------
// scatter GT connectivity into matches_gt, semantics targets, and record
// per-(b,m) -matches contribution for the BCE cross term (no fp atomics).
// pred is injective per batch -> all scatter destinations distinct.
// ---------------------------------------------------------------------------
__global__ void scatter_kernel(const int* __restrict__ pred, const int* __restrict__ pts_ins,
                               const int* __restrict__ pts_type, const float* __restrict__ matches,
                               float* __restrict__ out, float* __restrict__ msub) {
  int idx = blockIdx.x * blockDim.x + threadIdx.x;
  if (idx >= BB * MM) return;
  int b = idx / MM, m = idx % MM;
  int i = pred[idx];
  out[3 + (size_t)BB * NN * NN + (size_t)b * NN + i] = (float)pts_type[idx];
  if (m < MM - 1 && pts_ins[idx] == pts_ins[idx + 1]) {
    int j = pred[idx + 1];
    size_t base = 3 + (size_t)b * NN * NN;
    out[base + (size_t)i * NN + j] = 1.0f;
    out[base + (size_t)j * NN + i] = 1.0f;
    size_t mb = (size_t)b * NN * NN;
    msub[idx] = -(matches[mb + (size_t)i * NN + j] + matches[mb + (size_t)j * NN + i]);
  }
}

// ---------------------------------------------------------------------------
// NLL: sum log-prob at target classes -> block partials
// ---------------------------------------------------------------------------
__global__ __launch_bounds__(256)
void sem_kernel(const float* __restrict__ semantics, const float* __restrict__ out,
                double* __restrict__ partial) {
  __shared__ double sdata[256];
  int idx = blockIdx.x * blockDim.x + threadIdx.x;   // exactly B*N threads
  int b = idx / NN, n = idx % NN;
  int t = (int)out[3 + (size_t)BB * NN * NN + idx];
  double lp = (double)semantics[((size_t)b * (NCC + 1) + t) * NN + n];
  sdata[threadIdx.x] = lp;
  __syncthreads();
  for (int st = 128; st > 0; st >>= 1) {
    if (threadIdx.x < st) sdata[threadIdx.x] += sdata[threadIdx.x + st];
    __syncthreads();
  }
  if (threadIdx.x == 0) partial[blockIdx.x] = sdata[0];
}

// ---------------------------------------------------------------------------
// sum of per-(b,m) min distances -> block partials
// ---------------------------------------------------------------------------
__global__ __launch_bounds__(256)
void cdist_kernel(const unsigned* __restrict__ minBits, double* __restrict__ partial) {
  __shared__ double sdata[256];
  int idx = blockIdx.x * blockDim.x + threadIdx.x;   // exactly B*M threads
  sdata[threadIdx.x] = (double)__uint_as_float(minBits[idx]);
  __syncthreads();
  for (int st = 128; st > 0; st >>= 1) {
    if (threadIdx.x < st) sdata[threadIdx.x] += sdata[threadIdx.x + st];
    __syncthreads();
  }
  if (threadIdx.x == 0) partial[blockIdx.x] = sdata[0];
}

// ---------------------------------------------------------------------------
// final serial sums (fixed order -> deterministic) and the three scalars
// ---------------------------------------------------------------------------
__global__ void finalize_kernel(const double* __restrict__ psp, int nsp,
                                const double* __restrict__ psem, int nsem,
                                const double* __restrict__ pcd, int ncd,
                                const float* __restrict__ msub,
                                float* __restrict__ out) {
  double sp = 0.0, sem = 0.0, cd = 0.0, ms = 0.0;
  for (int i = 0; i < nsp; ++i) sp += psp[i];
  for (int i = 0; i < nsem; ++i) sem += psem[i];
  for (int i = 0; i < ncd; ++i) cd += pcd[i];
  for (int i = 0; i < BB * MM; ++i) ms += (double)msub[i];
  out[0] = (float)(cd / (double)(BB * MM));
  out[1] = (float)((sp + ms) / ((double)BB * NN * NN));
  out[2] = (float)(-sem / (double)(BB * NN));
}

// ---------------------------------------------------------------------------
extern "C" void kernel_launch(void* const* d_in, const int* in_sizes, int n_in,
                              void* d_out, int out_size, void* d_ws, size_t ws_size,
                              hipStream_t stream) {
  const float* matches   = (const float*)d_in[0];  // [B,N,N]
  const float* positions = (const float*)d_in[1];  // [B,N,2]
  const float* semantics = (const float*)d_in[2];  // [B,NC+1,N]
  // d_in[3] = masks (all ones, unused)
  const float* pts_gt    = (const float*)d_in[4];  // [B,M,2]
  const int*   pts_ins   = (const int*)d_in[5];    // [B,M]
  const int*   pts_type  = (const int*)d_in[6];    // [B,M]
  float* out = (float*)d_out;

  // workspace layout (~8.45 MB; cost matrix stays L2-resident on 192MB L2)
  char* ws = (char*)d_ws;
  size_t off = 0;
  float*    cost_t  = (float*)(ws + off);    off += (size_t)BB * MM * NN * sizeof(float);
  unsigned* minBits = (unsigned*)(ws + off); off += (size_t)BB * MM * sizeof(unsigned);
  int*      pred    = (int*)(ws + off);      off += (size_t)BB * MM * sizeof(int);
  double*   psp     = (double*)(ws + off);   off += 2048 * sizeof(double);
  double*   psem    = (double*)(ws + off);   off += 32 * sizeof(double);
  double*   pcd     = (double*)(ws + off);   off += 16 * sizeof(double);
  float*    msub    = (float*)(ws + off);    off += (size_t)BB * MM * sizeof(float);
  (void)in_sizes; (void)n_in; (void)out_size; (void)ws_size;

  init_ws_kernel<<<(BB * MM + 255) / 256, 256, 0, stream>>>(minBits, msub);

  dim3 cg(MM / 16, NN / 16, BB);                   // 16 x 32 x 16 waves
  cost_wmma_kernel<<<cg, 32, 0, stream>>>(positions, semantics, pts_gt, pts_type,
                                          cost_t, minBits);

  hungarian_kernel<<<BB, 256, 0, stream>>>(cost_t, pred);

  zero_out_kernel<<<4096, 256, 0, stream>>>(out);
  softplus_kernel<<<2048, 256, 0, stream>>>(matches, psp);
  scatter_kernel<<<(BB * MM + 255) / 256, 256, 0, stream>>>(pred, pts_ins, pts_type,
                                                            matches, out, msub);
  sem_kernel<<<(BB * NN) / 256, 256, 0, stream>>>(semantics, out, psem);
  cdist_kernel<<<(BB * MM) / 256, 256, 0, stream>>>(minBits, pcd);
  finalize_kernel<<<1, 1, 0, stream>>>(psp, 2048, psem, 32, pcd, 16, msub, out);
}